// RNNAcceptor_72748156059757
// MI455X (gfx1250) — compile-verified
//
#include <hip/hip_runtime.h>

#define B_    2048
#define T_    2048
#define E_    50
#define H_    50
#define G4_   200     // 4*H
#define GP_   208     // padded gate width (13 tiles of 16)
#define MT    16      // batch rows per workgroup
#define NW    13      // waves per workgroup: one per 16-col tile of 4H
#define CHUNK 256     // token staging chunk (time steps)

typedef __attribute__((ext_vector_type(2))) float v2f;
typedef __attribute__((ext_vector_type(8))) float v8f;

// ---------------------------------------------------------------------------
// Kernel 1: gate table  Gtab[v][r] = b_ih[r] + b_hh[r] + embed[v,:] . W_ih[r,:]
// (x-path of the LSTM collapses to a 13-entry lookup since vocab=13)
// ---------------------------------------------------------------------------
__global__ void gate_table_kernel(const float* __restrict__ embed,
                                  const float* __restrict__ W_ih,
                                  const float* __restrict__ b_ih,
                                  const float* __restrict__ b_hh,
                                  float* __restrict__ Gtab) {
  int idx = blockIdx.x * blockDim.x + threadIdx.x;
  if (idx >= 13 * G4_) return;
  int v = idx / G4_, r = idx % G4_;
  float s = b_ih[r] + b_hh[r];
#pragma unroll
  for (int e = 0; e < E_; ++e) s += embed[v * E_ + e] * W_ih[r * E_ + e];
  Gtab[idx] = s;
}

// fast tanh via v_exp_f32 + v_rcp_f32 (saturates correctly at +-1)
__device__ __forceinline__ float fast_tanh(float x) {
  return 1.0f - 2.0f * __builtin_amdgcn_rcpf(1.0f + __expf(2.0f * x));
}

// ---------------------------------------------------------------------------
// Kernel 2: the sequential LSTM. One workgroup = 16 batch rows, 13 waves.
// Wave w owns gate columns [16w, 16w+16). Recurrent GEMM hx@W_hh^T runs on
// V_WMMA_F32_16X16X4_F32 (f32 tensor op, preserves precision over 2048 steps).
// W_hh^T B-fragments live in 26 persistent VGPRs per lane.
// ---------------------------------------------------------------------------
__global__ __launch_bounds__(NW * 32) void lstm_kernel(
    const int* __restrict__ x, const float* __restrict__ Gtab_g,
    const float* __restrict__ W_hh,
    const float* __restrict__ W1, const float* __restrict__ b1,
    const float* __restrict__ W2, const float* __restrict__ b2,
    float* __restrict__ out) {
  __shared__ float sGtab[13 * GP_];    // 10.8 KB  gate table, padded stride
  __shared__ int   sTok[MT * CHUNK];   // 16 KB    token chunk
  __shared__ float sHx[MT * 52];       // hidden state, cols 50..51 kept zero
  __shared__ float sCx[MT * 52];       // cell state
  __shared__ float sGate[MT * GP_];    // activated gates: [m][gate*52 + j]

  const int tid  = threadIdx.x;
  const int wave = tid >> 5;           // 0..12 : which 16-col gate tile
  const int lane = tid & 31;
  const int nn   = lane & 15;          // N (and A-row M) within tile
  const int hi   = lane >> 4;          // half-wave selector for K striping
  const int row0 = blockIdx.x * MT;

  // stage gate table into padded layout; zero the pad columns
  for (int i = tid; i < 13 * GP_; i += NW * 32) {
    int v = i / GP_, r = i % GP_;
    sGtab[i] = (r < G4_) ? Gtab_g[v * G4_ + r] : 0.f;
  }
  for (int i = tid; i < MT * 52; i += NW * 32) { sHx[i] = 0.f; sCx[i] = 0.f; }

  // Preload B fragments of W_hh^T for this wave's 16 gate columns.
  // 32-bit B 4x16 striping: V0/V1 = K,K+1 (lanes 0-15) / K+2,K+3 (lanes 16-31).
  v2f Bf[13];
#pragma unroll
  for (int k = 0; k < 13; ++k) {
    int r  = wave * 16 + nn;     // gate-output row of W_hh (N of the GEMM)
    int kb = k * 4 + hi * 2;     // hidden index (K of the GEMM)
    float bx = 0.f, by = 0.f;
    if (r < G4_) {
      if (kb     < H_) bx = W_hh[r * H_ + kb];
      if (kb + 1 < H_) by = W_hh[r * H_ + kb + 1];
    }
    Bf[k].x = bx; Bf[k].y = by;
  }
  __syncthreads();

  const int col = wave * 16 + nn;      // global gate column (0..207)
  // branchless activation constants: a = c0 + c1 * rcp(1 + exp(s*x))
  // gate 0,1,3 -> sigmoid (s=-1, c1=1, c0=0); gate 2 -> tanh (s=2, c1=-2, c0=1)
  const int  gsel = col / 50;          // 0:i 1:f 2:g 3:o (pad cols -> 4, guarded)
  const bool isg  = (gsel == 2);
  const float sA  = isg ?  2.0f : -1.0f;
  const float c1A = isg ? -2.0f :  1.0f;
  const float c0A = isg ?  1.0f :  0.0f;
  const int  jjA  = col - gsel * 50;

  for (int tc0 = 0; tc0 < T_; tc0 += CHUNK) {
    // cooperatively stage the next 256 time steps of tokens for our 16 rows
    for (int i = tid; i < MT * CHUNK; i += NW * 32) {
      int m = i / CHUNK, tt = i % CHUNK;
      sTok[i] = x[(size_t)(row0 + m) * T_ + tc0 + tt];
    }
    if (tc0 + CHUNK < T_)
      __builtin_prefetch(&x[(size_t)row0 * T_ + tc0 + CHUNK], 0, 1);
    __syncthreads();

    for (int tt = 0; tt < CHUNK; ++tt) {
      // ---- accumulator init from gate table: x-path + both biases ----
      // unguarded: sGtab is padded to stride 208, pad cols feed discarded lanes
      int toks[8];
#pragma unroll
      for (int j = 0; j < 8; ++j) toks[j] = sTok[(j + hi * 8) * CHUNK + tt];
      v8f c;
#pragma unroll
      for (int j = 0; j < 8; ++j) c[j] = sGtab[toks[j] * GP_ + col];

      // ---- recurrent GEMM: c += hx(16x50) @ W_hh^T(50x16), K padded to 52 --
#pragma unroll
      for (int k = 0; k < 13; ++k) {
        v2f a;
        int kb = k * 4 + hi * 2;
        a.x = sHx[nn * 52 + kb];
        a.y = sHx[nn * 52 + kb + 1];
        c = __builtin_amdgcn_wmma_f32_16x16x4_f32(false, a, false, Bf[k],
                                                  (short)0, c, false, false);
      }
      // ---- activations, scatter into gate buffer ----
      if (col < G4_) {
#pragma unroll
        for (int j = 0; j < 8; ++j) {
          int m   = j + hi * 8;           // C/D layout: VGPR j -> M=j / j+8
          float t = __expf(sA * c[j]);
          float r = __builtin_amdgcn_rcpf(1.0f + t);
          sGate[m * GP_ + gsel * 52 + jjA] = c0A + c1A * r;
        }
      }
      __syncthreads();
      // ---- elementwise state update: c = f*c + i*g ; h = o*tanh(c) ----
      for (int i = tid; i < MT * H_; i += NW * 32) {
        int m = i / H_, j = i % H_;
        float gi = sGate[m * GP_ +       j];
        float gf = sGate[m * GP_ +  52 + j];
        float gg = sGate[m * GP_ + 104 + j];
        float go = sGate[m * GP_ + 156 + j];
        float cn = gf * sCx[m * 52 + j] + gi * gg;
        sCx[m * 52 + j] = cn;
        sHx[m * 52 + j] = go * fast_tanh(cn);
      }
      __syncthreads();
    }
  }

  // ---- epilogue MLP: h = relu(hx @ W1^T + b1); out = h @ W2^T + b2 ----
  for (int i = tid; i < MT * H_; i += NW * 32) {
    int m = i / H_, j = i % H_;
    float s = b1[j];
#pragma unroll
    for (int e = 0; e < H_; ++e) s += sHx[m * 52 + e] * W1[j * H_ + e];
    sGate[m * 52 + j] = fmaxf(s, 0.f);   // reuse sGate as h buffer
  }
  __syncthreads();
  if (tid < MT) {
    float s = b2[0];
#pragma unroll
    for (int e = 0; e < H_; ++e) s += sGate[tid * 52 + e] * W2[e];
    out[row0 + tid] = s;
  }
}

// ---------------------------------------------------------------------------
extern "C" void kernel_launch(void* const* d_in, const int* in_sizes, int n_in,
                              void* d_out, int out_size, void* d_ws,
                              size_t ws_size, hipStream_t stream) {
  const int*   x     = (const int*)  d_in[0];
  const float* embed = (const float*)d_in[1];
  const float* W_ih  = (const float*)d_in[2];
  const float* W_hh  = (const float*)d_in[3];
  const float* b_ih  = (const float*)d_in[4];
  const float* b_hh  = (const float*)d_in[5];
  const float* W1    = (const float*)d_in[6];
  const float* b1    = (const float*)d_in[7];
  const float* W2    = (const float*)d_in[8];
  const float* b2    = (const float*)d_in[9];
  float* out  = (float*)d_out;
  float* Gtab = (float*)d_ws;          // 13*200 floats of scratch

  gate_table_kernel<<<(13 * G4_ + 255) / 256, 256, 0, stream>>>(
      embed, W_ih, b_ih, b_hh, Gtab);
  lstm_kernel<<<B_ / MT, NW * 32, 0, stream>>>(
      x, Gtab, W_hh, W1, b1, W2, b2, out);
}